// LengthRegurator_13348758355985
// MI455X (gfx1250) — compile-verified
//
#include <hip/hip_runtime.h>
#include <hip/hip_bf16.h>
#include <stdint.h>

// Problem constants (match reference)
#define BATCH   32
#define CCH     384
#define T_TEXT  1024
#define T_FEAT  4096

// Native clang vector type: required by __builtin_nontemporal_store
typedef __attribute__((ext_vector_type(4))) float f32x4;
typedef __attribute__((ext_vector_type(4))) int   i32x4;

// ---------------------------------------------------------------------------
// Kernel 1: per-batch inclusive cumsum of durations w[b, 0..1023] -> cw
// One 256-thread block per batch row; each thread owns 4 elements (int4).
// ---------------------------------------------------------------------------
__global__ __launch_bounds__(256) void lr_scan(const int* __restrict__ w,
                                               int* __restrict__ cw) {
    __shared__ int sums[256];
    const int b   = blockIdx.x;
    const int tid = threadIdx.x;

    i32x4 v = ((const i32x4*)(w + b * T_TEXT))[tid];
    int s1 = v.x;
    int s2 = s1 + v.y;
    int s3 = s2 + v.z;
    int s4 = s3 + v.w;

    sums[tid] = s4;
    __syncthreads();

    // Hillis–Steele inclusive scan of per-thread totals
    int acc = s4;
    #pragma unroll
    for (int off = 1; off < 256; off <<= 1) {
        int t = (tid >= off) ? sums[tid - off] : 0;
        __syncthreads();
        acc += t;
        sums[tid] = acc;
        __syncthreads();
    }
    int excl = acc - s4;

    i32x4 o;
    o.x = excl + s1;
    o.y = excl + s2;
    o.z = excl + s3;
    o.w = excl + s4;
    ((i32x4*)(cw + b * T_TEXT))[tid] = o;
}

// ---------------------------------------------------------------------------
// Kernel 2: gather. Grid: (T_FEAT/1024, BATCH, CCH/64), 256 threads.
// - Stage cw[b,:] (4 KB) into LDS with gfx1250 async-to-LDS copy
//   (global_load_async_to_lds_b128, tracked by ASYNCcnt).
// - Each thread: 4 consecutive frames t -> binary search idx in LDS.
// - Loop 64 channels: 4 gathered x loads (x = 48 MiB stays resident in the
//   192 MB L2) and one coalesced 128-bit NON-TEMPORAL store (out = 192 MiB
//   write-once stream; NT keeps it from evicting x out of L2).
// ---------------------------------------------------------------------------
__global__ __launch_bounds__(256) void lr_gather(const float* __restrict__ x,
                                                 const float* __restrict__ x_mask,
                                                 const float* __restrict__ y_mask,
                                                 const int*   __restrict__ cw,
                                                 float*       __restrict__ out) {
    __shared__ int cw_lds[T_TEXT];

    const int b   = blockIdx.y;
    const int tid = threadIdx.x;

    // ---- async copy cw row (1024 ints = 4 KB) into LDS: 16 B per lane ----
    // Generic pointers into LDS carry the LDS byte offset in their low 32
    // bits (aperture in the high bits), so this yields a valid DS address.
    unsigned lds_addr = (unsigned)(uintptr_t)cw_lds + (unsigned)(tid * 16);
    unsigned voff     = (unsigned)(tid * 16);
    const int* src    = cw + b * T_TEXT;
    asm volatile("global_load_async_to_lds_b128 %0, %1, %2"
                 :
                 : "v"(lds_addr), "v"(voff), "s"(src)
                 : "memory");
    asm volatile("s_wait_asynccnt 0" ::: "memory");
    __syncthreads();

    const int total  = cw_lds[T_TEXT - 1];
    const int t_base = blockIdx.x * 1024 + tid * 4;

    int   idx[4];
    float scale[4];
    #pragma unroll
    for (int j = 0; j < 4; ++j) {
        const int t = t_base + j;
        // searchsorted(cw, t, side="right"): first i with cw[i] > t
        int lo = 0, hi = T_TEXT;
        #pragma unroll
        for (int step = 0; step < 10; ++step) {
            int mid = (lo + hi) >> 1;
            if (cw_lds[mid] <= t) lo = mid + 1; else hi = mid;
        }
        int id = lo < (T_TEXT - 1) ? lo : (T_TEXT - 1);
        idx[j] = id;
        float v = (t < total)
                    ? x_mask[b * T_TEXT + id] * y_mask[b * T_FEAT + t]
                    : 0.0f;
        scale[j] = v;
    }

    const int    c0 = blockIdx.z * 64;
    const float* xb = x   + (size_t)b * CCH * T_TEXT + (size_t)c0 * T_TEXT;
    float*       ob = out + (size_t)b * CCH * T_FEAT + (size_t)c0 * T_FEAT + t_base;

    #pragma unroll 4
    for (int c = 0; c < 64; ++c) {
        const float* xr = xb + (size_t)c * T_TEXT;
        f32x4 o;
        o.x = xr[idx[0]] * scale[0];
        o.y = xr[idx[1]] * scale[1];
        o.z = xr[idx[2]] * scale[2];
        o.w = xr[idx[3]] * scale[3];
        // Non-temporal 128-bit store: stream output past L2.
        __builtin_nontemporal_store(o, (f32x4*)(ob + (size_t)c * T_FEAT));
    }
}

// ---------------------------------------------------------------------------
// Host launcher
// ---------------------------------------------------------------------------
extern "C" void kernel_launch(void* const* d_in, const int* in_sizes, int n_in,
                              void* d_out, int out_size, void* d_ws, size_t ws_size,
                              hipStream_t stream) {
    const float* x      = (const float*)d_in[0];  // (B, C, T_TEXT) f32
    const int*   w      = (const int*)  d_in[1];  // (B, T_TEXT)    i32
    const float* x_mask = (const float*)d_in[2];  // (B, 1, T_TEXT) f32
    const float* y_mask = (const float*)d_in[3];  // (B, 1, T_FEAT) f32
    float*       out    = (float*)d_out;          // (B, C, T_FEAT) f32

    int* cw = (int*)d_ws;                         // BATCH*T_TEXT ints = 128 KB

    lr_scan<<<BATCH, 256, 0, stream>>>(w, cw);

    dim3 grid(T_FEAT / 1024, BATCH, CCH / 64);
    lr_gather<<<grid, 256, 0, stream>>>(x, x_mask, y_mask, cw, out);
}